// CTCLoss_11132555231245
// MI455X (gfx1250) — compile-verified
//
#include <hip/hip_runtime.h>
#include <stdint.h>

// Problem constants (match the reference).
#define T_DIM 2048
#define B_DIM 64
#define C_DIM 128
#define S_DIM 256
#define L_DIM (2 * S_DIM + 1)   // 513 extended labels
#define CHUNK 17                 // positions per lane: 32*17 = 544 >= 513
#define NEGV (-1e30f)
#define LOG2E 1.4426950408889634f
#define LN2   0.6931471805599453f

#if __has_builtin(__builtin_amdgcn_global_load_async_to_lds_b128) && \
    __has_builtin(__builtin_amdgcn_s_wait_asynccnt)
#define HAS_ASYNC 1
#else
#define HAS_ASYNC 0
#endif

typedef __attribute__((ext_vector_type(4))) float f4;
// Match the builtin's expected pointee type: 16-byte int vector.
typedef int gv4i __attribute__((vector_size(4 * sizeof(int))));
typedef __attribute__((address_space(1))) gv4i* g4p;  // global
typedef __attribute__((address_space(3))) gv4i* l4p;  // LDS

__device__ __forceinline__ void prefetch16(const float* g, float* l) {
#if HAS_ASYNC
  // Async DMA global -> LDS, tracked with ASYNCcnt (CDNA5 path).
  __builtin_amdgcn_global_load_async_to_lds_b128(
      (g4p)const_cast<float*>(g), (l4p)l, /*imm offset*/ 0, /*cpol*/ 0);
#else
  *(f4*)l = *(const f4*)g;   // synchronous fallback
#endif
}

#if HAS_ASYNC
#define WAIT_ASYNC(n)                       \
  do {                                      \
    __builtin_amdgcn_s_wait_asynccnt(n);    \
    asm volatile("" ::: "memory");          \
  } while (0)
#else
#define WAIT_ASYNC(n) asm volatile("" ::: "memory")
#endif

// Drain wave's LDS reads before the async engine may overwrite a buffer.
#define DS_DRAIN() asm volatile("s_wait_dscnt 0x0" ::: "memory")

__global__ __launch_bounds__(32) void ctc_alpha_kernel(
    const float* __restrict__ log_probs,   // [T,B,C]
    const int* __restrict__ targets,       // [B,S]
    const int* __restrict__ input_lengths, // [B]
    const int* __restrict__ target_lengths,// [B]
    float* __restrict__ loss_ws) {         // [B]
  __shared__ float s_lp[2 * C_DIM];        // double-buffered class row
  __shared__ float s_alpha[32 * CHUNK];    // final alpha spill for extraction

  const int b = blockIdx.x;
  const int lane = threadIdx.x;            // wave32
  const int base = lane * CHUNK;
  const int ilen = input_lengths[b];
  const int tlen = target_lengths[b];

  // ---- per-lane extended-label table + skip bias (registers) ----
  // bias[i] = 0 if the l-2 "skip" path is allowed at position p, else NEGV.
  // a2 + bias gives a2 (skip) or ~-2e30 (no skip) -> exp2 underflows to 0.
  int   lab[CHUNK];
  float bias[CHUNK];
#pragma unroll
  for (int i = 0; i < CHUNK; ++i) {
    int p = base + i;
    int l = 0;
    float bi = NEGV;
    if (p < L_DIM && (p & 1)) {
      int j = (p - 1) >> 1;
      l = targets[b * S_DIM + j];
      if (p >= 3) {
        int l2 = targets[b * S_DIM + j - 1];
        if (l != l2) bi = 0.0f;  // skip transition allowed
      }
    }
    lab[i]  = l;   // even positions & out-of-range -> blank(0)
    bias[i] = bi;
  }

#define ISSUE_PREFETCH(tt, buf)                                               \
  do {                                                                        \
    const float* _g =                                                         \
        log_probs + ((size_t)(tt) * B_DIM + (size_t)b) * C_DIM + lane * 4;    \
    float* _l = &s_lp[(buf) * C_DIM + lane * 4];                              \
    prefetch16(_g, _l);                                                       \
  } while (0)

  // ---- t = 0: fetch row, init alpha (log2 domain: beta = alpha * log2e) ----
  ISSUE_PREFETCH(0, 0);
  WAIT_ASYNC(0);

  float alpha[CHUNK];
#pragma unroll
  for (int i = 0; i < CHUNK; ++i) {
    int p = base + i;
    float v = NEGV;
    if (p <= 1) v = s_lp[lab[i]] * LOG2E;  // p=0: blank lp; p=1: first label lp
    alpha[i] = v;
  }

  DS_DRAIN();
  ISSUE_PREFETCH(1, 1);

  // ---- forward scan over time ----
#pragma unroll 2
  for (int t = 1; t < T_DIM; ++t) {
    int nt = (t + 1 < T_DIM) ? (t + 1) : (T_DIM - 1);  // clamp (redundant tail fetch)
    DS_DRAIN();                        // prior reads of the recycled buffer done
    ISSUE_PREFETCH(nt, (t + 1) & 1);   // overlap next row with this step
    WAIT_ASYNC(1);                     // current row resident; newest still in flight

    if (t < ilen) {                    // uniform (scalar) branch
      const float* lpc = &s_lp[(t & 1) * C_DIM];

      // boundary values from the previous lane (old alpha)
      float sh16 = __shfl_up(alpha[CHUNK - 1], 1, 32);
      float sh15 = __shfl_up(alpha[CHUNK - 2], 1, 32);
      if (lane == 0) { sh16 = NEGV; sh15 = NEGV; }

      // descending order -> in-place update, no alpha copy needed
#pragma unroll
      for (int i = CHUNK - 1; i >= 0; --i) {
        float a0 = alpha[i];
        float a1 = (i >= 1) ? alpha[i - 1] : sh16;
        float a2 = (i >= 2) ? alpha[i - 2] : ((i == 1) ? sh16 : sh15);
        float a2m = a2 + bias[i];                      // skip gate (additive)
        float m = fmaxf(fmaxf(a0, a1), a2m);           // v_max3_f32
        float s = __builtin_amdgcn_exp2f(a0 - m) +
                  __builtin_amdgcn_exp2f(a1 - m) +
                  __builtin_amdgcn_exp2f(a2m - m);     // s >= 1
        float merged = m + __builtin_amdgcn_logf(s);   // log2 domain
        alpha[i] = fmaf(lpc[lab[i]], LOG2E, merged);   // + lp_t * log2e
      }
    }
  }

  // ---- extract alpha[2*tlen], alpha[2*tlen-1] ----
  __syncthreads();
#pragma unroll
  for (int i = 0; i < CHUNK; ++i) s_alpha[base + i] = alpha[i];
  __syncthreads();

  if (lane == 0) {
    float ab = s_alpha[2 * tlen];
    float al = s_alpha[2 * tlen - 1];
    float m = fmaxf(ab, al);
    float ll2 = m + __builtin_amdgcn_logf(__builtin_amdgcn_exp2f(ab - m) +
                                          __builtin_amdgcn_exp2f(al - m));
    float ll = ll2 * LN2;  // back to natural log
    loss_ws[b] = -ll / (float)tlen;
  }
}

__global__ void ctc_reduce_kernel(const float* __restrict__ ws,
                                  float* __restrict__ out) {
  if (threadIdx.x == 0 && blockIdx.x == 0) {
    float s = 0.0f;
    for (int i = 0; i < B_DIM; ++i) s += ws[i];
    out[0] = s / (float)B_DIM;
  }
}

extern "C" void kernel_launch(void* const* d_in, const int* in_sizes, int n_in,
                              void* d_out, int out_size, void* d_ws, size_t ws_size,
                              hipStream_t stream) {
  const float* log_probs      = (const float*)d_in[0];
  const int*   targets        = (const int*)d_in[1];
  const int*   input_lengths  = (const int*)d_in[2];
  const int*   target_lengths = (const int*)d_in[3];
  float* out = (float*)d_out;
  float* ws  = (float*)d_ws;  // needs B_DIM floats

  ctc_alpha_kernel<<<dim3(B_DIM), dim3(32), 0, stream>>>(
      log_probs, targets, input_lengths, target_lengths, ws);
  ctc_reduce_kernel<<<dim3(1), dim3(32), 0, stream>>>(ws, out);
}